// LinearEncoder_1546188226766
// MI455X (gfx1250) — compile-verified
//
#include <hip/hip_runtime.h>
#include <hip/hip_bf16.h>

typedef __attribute__((ext_vector_type(2))) float v2f;
typedef __attribute__((ext_vector_type(4))) float v4f;
typedef __attribute__((ext_vector_type(8))) float v8f;

#define NN      512   // nodes
#define NIN     256   // per-node feature dim (K of each half-GEMM)
#define TWO_IN  512   // W row length

// ---------------------------------------------------------------------------
// Kernel 1: AB[m, n]  (512 x 512, row-major), m = node row.
//   n in [0,256)   : AB[m,n]       = sum_k x[m,k] * W[n,       k]       (A half)
//   n in [256,512) : AB[m,n]       = sum_k x[m,k] * W[n-256, 256+k]     (B half)
// One wave computes one 16x16 tile with V_WMMA_F32_16X16X4_F32 (K-step 4).
// ---------------------------------------------------------------------------
__global__ void gemm_wmma(const float* __restrict__ x,
                          const float* __restrict__ W,
                          float* __restrict__ AB) {
    const int wave = threadIdx.x >> 5;
    const int lane = threadIdx.x & 31;
    const int tile = blockIdx.x * (blockDim.x >> 5) + wave;  // 0..1023
    const int mt = tile >> 5;            // 0..31  (M tile)
    const int nt = tile & 31;            // 0..31  (N tile)
    const int m0 = mt << 4;
    const int n0 = nt << 4;
    const int half = n0 >> 8;            // 0 -> W1 half, 1 -> W2 half
    const int lr = lane & 15;            // row/col within tile
    const int lh = lane >> 4;            // K sub-phase (0 or 1)

    // A operand: per-lane (M = m0+lr, K = k0 + 2*lh + {0,1}) -> contiguous pair
    const float* arow = x + (m0 + lr) * NIN + 2 * lh;
    // B operand: per-lane (K = k0 + 2*lh + {0,1}, N = n0+lr); B[k][n] = W[nLocal, half*256 + k]
    const float* brow = W + ((n0 & 255) + lr) * TWO_IN + half * 256 + 2 * lh;

    v8f c = {0.f, 0.f, 0.f, 0.f, 0.f, 0.f, 0.f, 0.f};
#pragma unroll 8
    for (int k0 = 0; k0 < NIN; k0 += 4) {
        v2f a = *(const v2f*)(arow + k0);
        v2f b = *(const v2f*)(brow + k0);
        // 8 args: (neg_a, A, neg_b, B, c_mod, C, reuse_a, reuse_b)
        c = __builtin_amdgcn_wmma_f32_16x16x4_f32(false, a, false, b,
                                                  (short)0, c, false, false);
    }

    // D layout: VGPR r -> row m0 + r + 8*lh, col n0 + lr
    float* dst = AB + (m0 + 8 * lh) * 512 + n0 + lr;
#pragma unroll
    for (int r = 0; r < 8; ++r)
        dst[r * 512] = c[r];
}

// ---------------------------------------------------------------------------
// Kernel 2: out[i,j,c] = (i==j) ? 0 : AB[min(i,j), c] + AB[max(i,j), 256+c] + b[c]
// Pure store-bandwidth kernel: 256 MB of NT float4 stores; reads hit L1/L2
// (AB = 1 MB, bias = 1 KB).
// ---------------------------------------------------------------------------
__global__ void broadcast_out(const float* __restrict__ AB,
                              const float* __restrict__ bias,
                              float* __restrict__ out) {
    const long long idx =
        (long long)blockIdx.x * blockDim.x + threadIdx.x;   // float4 index
    const int c = (int)(idx & 63) << 2;                     // channel base
    const int j = (int)((idx >> 6) & 511);
    const int i = (int)(idx >> 15);

    v4f r;
    if (i == j) {
        r = (v4f){0.f, 0.f, 0.f, 0.f};
    } else {
        const int lo = i < j ? i : j;
        const int hi = i < j ? j : i;
        const v4f a  = *(const v4f*)(AB + lo * 512 + c);
        const v4f b  = *(const v4f*)(AB + hi * 512 + 256 + c);
        const v4f bb = *(const v4f*)(bias + c);
        r = a + b + bb;
    }
    __builtin_nontemporal_store(r, (v4f*)out + idx);
}

// ---------------------------------------------------------------------------
extern "C" void kernel_launch(void* const* d_in, const int* in_sizes, int n_in,
                              void* d_out, int out_size, void* d_ws, size_t ws_size,
                              hipStream_t stream) {
    const float* x = (const float*)d_in[0];   // [512, 256]
    const float* W = (const float*)d_in[1];   // [256, 512]
    const float* b = (const float*)d_in[2];   // [256]
    float* out = (float*)d_out;               // [512, 512, 256]
    float* AB  = (float*)d_ws;                // [512, 512] = 1 MB scratch

    // 1024 tiles, 8 waves (256 threads) per block -> 128 blocks
    gemm_wmma<<<128, 256, 0, stream>>>(x, W, AB);

    // 512*512*64 float4 elements / 256 threads = 65536 blocks
    broadcast_out<<<65536, 256, 0, stream>>>(AB, b, out);
}